// BinarySegmentationLoss_7507602833962
// MI455X (gfx1250) — compile-verified
//
#include <hip/hip_runtime.h>
#include <hip/hip_bf16.h>
#include <math.h>

typedef __attribute__((ext_vector_type(2))) float v2f;
typedef __attribute__((ext_vector_type(8))) float v8f;

#define N_IMG   8
#define HW      65536           // 256*256
#define TOTAL   (N_IMG * HW)    // 524288
#define INFV    1.0e10f
#define EPSV    1.0e-6f
#define MAXD2   130050.0f       // (255^2 + 255^2)

// ---------------------------------------------------------------------------
// Kernel 1: per-row 1-D squared distance to nearest fg / nearest bg pixel.
// One block per (image, row); thread = px. Brute force over qx with the
// thresholded row in LDS (broadcast reads, no bank conflicts).
// ---------------------------------------------------------------------------
__global__ void edt_rows_kernel(const float* __restrict__ targ,
                                float* __restrict__ rfg,
                                float* __restrict__ rbg) {
    __shared__ float sfg[256];
    const int n  = blockIdx.x >> 8;
    const int y  = blockIdx.x & 255;
    const int px = threadIdx.x;
    const int rowbase = n * HW + y * 256;

    sfg[px] = (targ[rowbase + px] > 0.5f) ? 1.0f : 0.0f;
    __syncthreads();

    float mf = INFV;   // min (px-qx)^2 over fg qx
    float mb = INFV;   // min (px-qx)^2 over bg qx
    #pragma unroll 8
    for (int qx = 0; qx < 256; ++qx) {
        const float dx = (float)(px - qx);
        const float d2 = dx * dx;
        const bool f = sfg[qx] > 0.5f;
        mf = f ? fminf(mf, d2) : mf;
        mb = f ? mb : fminf(mb, d2);
    }
    rfg[rowbase + px] = mf;
    rbg[rowbase + px] = mb;
}

// ---------------------------------------------------------------------------
// Kernel 2: column min-plus combine d[py,px] = min_qy (py-qy)^2 + d1[qy,px],
// empty-foreground fallback, signed sqrt -> phi.
// One block per (image, py); thread = px (coalesced across qy rows, rows hit
// in L2/WGP$ on the 256x re-reads).
// ---------------------------------------------------------------------------
__global__ void edt_cols_kernel(const float* __restrict__ targ,
                                const float* __restrict__ rfg,
                                const float* __restrict__ rbg,
                                float* __restrict__ phi) {
    const int n  = blockIdx.x >> 8;
    const int py = blockIdx.x & 255;
    const int px = threadIdx.x;
    const float* rf = rfg + n * HW;
    const float* rb = rbg + n * HW;

    // warm GL2 for this column walk (CDNA5 global_prefetch_b8)
    __builtin_prefetch(rf + px, 0, 1);
    __builtin_prefetch(rb + px, 0, 1);

    float df = 1.0e30f;
    float db = 1.0e30f;
    #pragma unroll 8
    for (int qy = 0; qy < 256; ++qy) {
        const float dy  = (float)(py - qy);
        const float dy2 = dy * dy;
        df = fminf(df, dy2 + rf[qy * 256 + px]);
        db = fminf(db, dy2 + rb[qy * 256 + px]);
    }
    if (df > 0.5f * INFV) df = MAXD2;
    if (db > 0.5f * INFV) db = MAXD2;

    const int idx = n * HW + py * 256 + px;
    const bool fg = targ[idx] > 0.5f;
    phi[idx] = fg ? -sqrtf(db) : sqrtf(df);
}

// ---------------------------------------------------------------------------
// Kernel 3: elementwise loss terms + WMMA f32 16x16x4 reductions.
// Each wave streams 64 values/iteration into the A operand (2 VGPRs/lane),
// B = ones(4x16), D accumulates across iterations (exact f32 MACs).
// 5 accumulators: sum(p*t), sum(p), sum(t), sum(phi*p), sum(focal).
// Block = 256 threads = 8 waves; 16 blocks per image, 4096 elems per block.
// ---------------------------------------------------------------------------
#define BLOCKS_PER_IMG 16
#define CHUNK          (HW / BLOCKS_PER_IMG)   // 4096
#define WAVE_ELEMS     (CHUNK / 8)             // 512
#define ITERS          (WAVE_ELEMS / 64)       // 8

__device__ __forceinline__ v8f wmma_acc(v2f a, v2f ones, v8f c) {
    // D = A(16x4 f32) x B(4x16 ones) + C ; exact f32 accumulation
    return __builtin_amdgcn_wmma_f32_16x16x4_f32(
        /*neg_a=*/false, a, /*neg_b=*/false, ones,
        /*c_mod=*/(short)0, c, /*reuse_a=*/false, /*reuse_b=*/false);
}

__device__ __forceinline__ float wave_total(v8f c) {
    // Each D row-sum is replicated across 16 columns; sum everything, /16.
    float s = c[0] + c[1] + c[2] + c[3] + c[4] + c[5] + c[6] + c[7];
    #pragma unroll
    for (int m = 16; m >= 1; m >>= 1) s += __shfl_xor(s, m, 32);
    return s * 0.0625f;
}

__global__ void reduce_kernel(const float* __restrict__ pred,
                              const float* __restrict__ targ,
                              const float* __restrict__ phi,
                              float* __restrict__ partials) {
    const int w    = threadIdx.x >> 5;
    const int lane = threadIdx.x & 31;
    const long base = (long)blockIdx.x * CHUNK + (long)w * WAVE_ELEMS + lane * 2;

    const v2f ones = {1.0f, 1.0f};
    v8f c_pt = {}, c_p = {}, c_t = {}, c_bp = {}, c_f = {};

    for (int it = 0; it < ITERS; ++it) {
        const long i = base + (long)it * 64;
        const float x0 = pred[i], x1 = pred[i + 1];
        const float t0 = targ[i], t1 = targ[i + 1];
        const float h0 = phi[i],  h1 = phi[i + 1];

        const float p0 = 1.0f / (1.0f + expf(-x0));
        const float p1 = 1.0f / (1.0f + expf(-x1));

        // focal terms (gamma = 2 -> square; alpha = 0.25)
        const float pc0 = fminf(fmaxf(p0, EPSV), 1.0f - EPSV);
        const float pc1 = fminf(fmaxf(p1, EPSV), 1.0f - EPSV);
        const float pt0 = pc0 * t0 + (1.0f - pc0) * (1.0f - t0);
        const float pt1 = pc1 * t1 + (1.0f - pc1) * (1.0f - t1);
        const float at0 = 0.25f * t0 + 0.75f * (1.0f - t0);
        const float at1 = 0.25f * t1 + 0.75f * (1.0f - t1);
        const float f0 = -at0 * (1.0f - pt0) * (1.0f - pt0) * logf(pt0);
        const float f1 = -at1 * (1.0f - pt1) * (1.0f - pt1) * logf(pt1);

        v2f a;
        a.x = p0 * t0; a.y = p1 * t1;  c_pt = wmma_acc(a, ones, c_pt);
        a.x = p0;      a.y = p1;       c_p  = wmma_acc(a, ones, c_p);
        a.x = t0;      a.y = t1;       c_t  = wmma_acc(a, ones, c_t);
        a.x = h0 * p0; a.y = h1 * p1;  c_bp = wmma_acc(a, ones, c_bp);
        a.x = f0;      a.y = f1;       c_f  = wmma_acc(a, ones, c_f);
    }

    __shared__ float ss[8][5];
    const float s_pt = wave_total(c_pt);
    const float s_p  = wave_total(c_p);
    const float s_t  = wave_total(c_t);
    const float s_bp = wave_total(c_bp);
    const float s_f  = wave_total(c_f);
    if (lane == 0) {
        ss[w][0] = s_pt; ss[w][1] = s_p; ss[w][2] = s_t;
        ss[w][3] = s_bp; ss[w][4] = s_f;
    }
    __syncthreads();
    if (threadIdx.x == 0) {
        float a0 = 0, a1 = 0, a2 = 0, a3 = 0, a4 = 0;
        for (int k = 0; k < 8; ++k) {
            a0 += ss[k][0]; a1 += ss[k][1]; a2 += ss[k][2];
            a3 += ss[k][3]; a4 += ss[k][4];
        }
        float* q = partials + (long)blockIdx.x * 5;
        q[0] = a0; q[1] = a1; q[2] = a2; q[3] = a3; q[4] = a4;
    }
}

// ---------------------------------------------------------------------------
// Kernel 4: deterministic serial combine of 128 block partials.
// ---------------------------------------------------------------------------
__global__ void final_kernel(const float* __restrict__ partials,
                             float* __restrict__ out) {
    if (threadIdx.x != 0 || blockIdx.x != 0) return;

    float dice_acc = 0.0f;
    for (int n = 0; n < N_IMG; ++n) {
        float inter = 0.0f, ps = 0.0f, ts = 0.0f;
        for (int b = 0; b < BLOCKS_PER_IMG; ++b) {
            const float* q = partials + (long)(n * BLOCKS_PER_IMG + b) * 5;
            inter += q[0]; ps += q[1]; ts += q[2];
        }
        dice_acc += (2.0f * inter + EPSV) / (ps + ts + EPSV);
    }
    const float dice_val = 1.0f - dice_acc * (1.0f / (float)N_IMG);

    float bsum = 0.0f, fsum = 0.0f;
    for (int b = 0; b < N_IMG * BLOCKS_PER_IMG; ++b) {
        bsum += partials[(long)b * 5 + 3];
        fsum += partials[(long)b * 5 + 4];
    }
    const float inv = 1.0f / (float)TOTAL;
    const float boundary_val = bsum * inv;
    const float focal_val    = fsum * inv;

    out[0] = dice_val + boundary_val + focal_val;  // loss
    out[1] = dice_val;
    out[2] = boundary_val;
    out[3] = focal_val;
}

// ---------------------------------------------------------------------------
extern "C" void kernel_launch(void* const* d_in, const int* in_sizes, int n_in,
                              void* d_out, int out_size, void* d_ws, size_t ws_size,
                              hipStream_t stream) {
    const float* pred = (const float*)d_in[0];
    const float* targ = (const float*)d_in[1];
    float* out = (float*)d_out;

    float* ws       = (float*)d_ws;
    float* rfg      = ws;                 // [TOTAL]
    float* rbg      = rfg + TOTAL;        // [TOTAL]
    float* phi      = rbg + TOTAL;        // [TOTAL]
    float* partials = phi + TOTAL;        // [128 * 5]

    edt_rows_kernel<<<N_IMG * 256, 256, 0, stream>>>(targ, rfg, rbg);
    edt_cols_kernel<<<N_IMG * 256, 256, 0, stream>>>(targ, rfg, rbg, phi);
    reduce_kernel<<<N_IMG * BLOCKS_PER_IMG, 256, 0, stream>>>(pred, targ, phi, partials);
    final_kernel<<<1, 32, 0, stream>>>(partials, out);
}